// Gemma3Base_25477746000395
// MI455X (gfx1250) — compile-verified
//
#include <hip/hip_runtime.h>
#include <hip/hip_bf16.h>
#include <math.h>

// ---------------------------------------------------------------------------
// Gemma3 prefill for MI455X (gfx1250, wave32).
// GEMMs run on the WMMA matrix pipe in exact fp32 via V_WMMA_F32_16X16X4_F32.
// All GEMM shapes are compile-time template constants so B-matrix strided
// addressing folds into immediate offsets (one base bump per K-slab), and
// each wave register-blocks a 32x32 C region (2x2 WMMA tiles) to amortize
// fragment loads: 4 WMMAs per 2 A-pair + 2 B-pair loads.
// Weights (~77MB/layer) fit the 192MB L2; HBM floor ~2GB / 86us @ 23.3TB/s.
// ---------------------------------------------------------------------------

#define LAYERS 26
#define D      1152
#define NH     4
#define HD     256
#define FFD    4608
#define TQ     512
#define CACHE_ 512
#define SK     1024        // CACHE + T
#define QKV_N  1536        // (NH + 2) * HD  (NG=1)
#define AO     1024        // NH * HD

typedef __attribute__((ext_vector_type(2))) float v2f;
typedef __attribute__((ext_vector_type(8))) float v8f;

#ifndef __has_builtin
#define __has_builtin(x) 0
#endif

// Real builtin only in the device pass (probe-verified on gfx1250:
// signature (bool, v2f, bool, v2f, short, v8f, bool, bool), lowers to
// v_wmma_f32_16x16x4_f32).  Host stub only needs to parse.
#if defined(__HIP_DEVICE_COMPILE__) && __has_builtin(__builtin_amdgcn_wmma_f32_16x16x4_f32)
#define WMMA_F32X4(a, b, c) \
    __builtin_amdgcn_wmma_f32_16x16x4_f32(false, (a), false, (b), (short)0, (c), false, false)
#else
#define WMMA_F32X4(a, b, c) (c)
#endif

// ---------------------------------------------------------------------------
// C = A(MxK, row-major) * B(KxN, row-major).  Compile-time K/LDA/LDB/LDC.
// Block = 256 threads = 8 waves laid out 2(M) x 4(N); block tile 64x128.
// Wave tile 32x32 = 2x2 WMMA tiles (32 accumulator VGPRs).
// Fragment layouts per ISA 7.12.2:
//   A 16x4 : lane r=lane&15 -> row, VGPR pair holds K={2h,2h+1}, h=lane>>4
//   B 4x16 : mirrored (lane -> col, VGPR pair holds same K split)
//   C 16x16: VGPR j -> row j+8h, col r
// blockIdx.z batches (per-head attention GEMMs).
// ---------------------------------------------------------------------------
template <int KDIM, int LDA, int LDB, int LDC>
__global__ __launch_bounds__(256) void gemm_nn_t(
    const float* __restrict__ A, const float* __restrict__ B, float* __restrict__ C,
    long sA, long sB, long sC)
{
    A += (long)blockIdx.z * sA;
    B += (long)blockIdx.z * sB;
    C += (long)blockIdx.z * sC;
    const int wave = threadIdx.x >> 5;
    const int wm   = wave >> 2;                 // 0..1
    const int wn   = wave & 3;                  // 0..3
    const int lane = threadIdx.x & 31;
    const int r    = lane & 15;
    const int hh   = lane >> 4;
    const int m0   = blockIdx.y * 64 + wm * 32;
    const int n0   = blockIdx.x * 128 + wn * 32;
    const float* Ar0 = A + (long)(m0 + r) * LDA;
    const float* Ar1 = Ar0 + 16 * LDA;
    const float* Bc  = B + (long)(2 * hh) * LDB + n0 + r;  // lane K-split folded in
    v8f c00 = {}, c01 = {}, c10 = {}, c11 = {};
    const int koff = 2 * hh;
    for (int k0 = 0; k0 < KDIM; k0 += 16) {
#pragma unroll
        for (int kk = 0; kk < 16; kk += 4) {
            const int k = k0 + kk + koff;       // A-side index
            const long bk = (long)(k0 + kk) * LDB;  // B-side: lane split in Bc
            v2f a0, a1, b0, b1;
            a0.x = Ar0[k];  a0.y = Ar0[k + 1];
            a1.x = Ar1[k];  a1.y = Ar1[k + 1];
            b0.x = Bc[bk];          b0.y = Bc[bk + LDB];
            b1.x = Bc[bk + 16];     b1.y = Bc[bk + LDB + 16];
            c00 = WMMA_F32X4(a0, b0, c00);
            c01 = WMMA_F32X4(a0, b1, c01);
            c10 = WMMA_F32X4(a1, b0, c10);
            c11 = WMMA_F32X4(a1, b1, c11);
        }
    }
    float* Cp = C + (long)(m0 + hh * 8) * LDC + n0 + r;
#pragma unroll
    for (int j = 0; j < 8; ++j) {
        Cp[(long)j * LDC]             = c00[j];
        Cp[(long)j * LDC + 16]        = c01[j];
        Cp[(long)(j + 16) * LDC]      = c10[j];
        Cp[(long)(j + 16) * LDC + 16] = c11[j];
    }
}

// C = A(MxK, row-major) * B^T, B stored row-major NxK (contiguous K pairs ->
// b64 loads).  Used for q@k_full^T and probs@v_full^T.
template <int KDIM, int LDA, int LDB, int LDC>
__global__ __launch_bounds__(256) void gemm_nt_t(
    const float* __restrict__ A, const float* __restrict__ B, float* __restrict__ C,
    long sA, long sB, long sC)
{
    A += (long)blockIdx.z * sA;
    B += (long)blockIdx.z * sB;
    C += (long)blockIdx.z * sC;
    const int wave = threadIdx.x >> 5;
    const int wm   = wave >> 2;
    const int wn   = wave & 3;
    const int lane = threadIdx.x & 31;
    const int r    = lane & 15;
    const int hh   = lane >> 4;
    const int m0   = blockIdx.y * 64 + wm * 32;
    const int n0   = blockIdx.x * 128 + wn * 32;
    const float* Ar0 = A + (long)(m0 + r) * LDA;
    const float* Ar1 = Ar0 + 16 * LDA;
    const float* Br0 = B + (long)(n0 + r) * LDB;
    const float* Br1 = Br0 + 16 * LDB;
    v8f c00 = {}, c01 = {}, c10 = {}, c11 = {};
    const int koff = 2 * hh;
    for (int k0 = 0; k0 < KDIM; k0 += 16) {
#pragma unroll
        for (int kk = 0; kk < 16; kk += 4) {
            const int k = k0 + kk + koff;
            v2f a0, a1, b0, b1;
            a0.x = Ar0[k];  a0.y = Ar0[k + 1];
            a1.x = Ar1[k];  a1.y = Ar1[k + 1];
            b0.x = Br0[k];  b0.y = Br0[k + 1];
            b1.x = Br1[k];  b1.y = Br1[k + 1];
            c00 = WMMA_F32X4(a0, b0, c00);
            c01 = WMMA_F32X4(a0, b1, c01);
            c10 = WMMA_F32X4(a1, b0, c10);
            c11 = WMMA_F32X4(a1, b1, c11);
        }
    }
    float* Cp = C + (long)(m0 + hh * 8) * LDC + n0 + r;
#pragma unroll
    for (int j = 0; j < 8; ++j) {
        Cp[(long)j * LDC]             = c00[j];
        Cp[(long)j * LDC + 16]        = c01[j];
        Cp[(long)(j + 16) * LDC]      = c10[j];
        Cp[(long)(j + 16) * LDC + 16] = c11[j];
    }
}

// ---------------------------------------------------------------------------
// Block-wide reductions (blockDim.x == 256)
// ---------------------------------------------------------------------------
__device__ __forceinline__ float block_reduce_sum(float v, float* red)
{
    const int tid = threadIdx.x;
    red[tid] = v;
    __syncthreads();
    for (int s = 128; s > 0; s >>= 1) {
        if (tid < s) red[tid] += red[tid + s];
        __syncthreads();
    }
    const float r = red[0];
    __syncthreads();
    return r;
}

__device__ __forceinline__ float block_reduce_max(float v, float* red)
{
    const int tid = threadIdx.x;
    red[tid] = v;
    __syncthreads();
    for (int s = 128; s > 0; s >>= 1) {
        if (tid < s) red[tid] = fmaxf(red[tid], red[tid + s]);
        __syncthreads();
    }
    const float r = red[0];
    __syncthreads();
    return r;
}

// out = rms_norm(x, w); one block per row
__global__ __launch_bounds__(256) void rmsnorm_kernel(
    const float* __restrict__ x, const float* __restrict__ w,
    float* __restrict__ out, int dim)
{
    __shared__ float red[256];
    const long row = blockIdx.x;
    const float* xr = x + row * dim;
    float ss = 0.f;
    for (int d = threadIdx.x; d < dim; d += 256) { const float v = xr[d]; ss += v * v; }
    const float inv = rsqrtf(block_reduce_sum(ss, red) / (float)dim + 1e-6f);
    float* orow = out + row * dim;
    for (int d = threadIdx.x; d < dim; d += 256)
        orow[d] = xr[d] * inv * (1.f + w[d]);
}

// h += rms_norm(y, w); one block per row
__global__ __launch_bounds__(256) void add_rmsnorm_kernel(
    float* __restrict__ h, const float* __restrict__ y,
    const float* __restrict__ w, int dim)
{
    __shared__ float red[256];
    const long row = blockIdx.x;
    const float* yr = y + row * dim;
    float ss = 0.f;
    for (int d = threadIdx.x; d < dim; d += 256) { const float v = yr[d]; ss += v * v; }
    const float inv = rsqrtf(block_reduce_sum(ss, red) / (float)dim + 1e-6f);
    float* hr = h + row * dim;
    for (int d = threadIdx.x; d < dim; d += 256)
        hr[d] += yr[d] * inv * (1.f + w[d]);
}

// strided row copy (cache -> k_full / v_full)
__global__ __launch_bounds__(256) void copy_rows_kernel(
    const float* __restrict__ src, float* __restrict__ dst,
    int cols, int src_ld, int dst_ld)
{
    const long r = blockIdx.x;
    for (int c = threadIdx.x; c < cols; c += 256)
        dst[r * dst_ld + c] = src[r * src_ld + c];
}

// Per query position: split qkv row, q/k rms-norm, RoPE, scatter to
// q[NH,T,HD], k_full[S,HD], v_full[HD,S] and the k/v output slices.
__global__ __launch_bounds__(256) void qkv_post_kernel(
    const float* __restrict__ qkv,
    const float* __restrict__ qw, const float* __restrict__ kw,
    const float* __restrict__ cosT, const float* __restrict__ sinT,
    float* __restrict__ qout, float* __restrict__ kfull, float* __restrict__ vfull,
    float* __restrict__ kslice, float* __restrict__ vslice)
{
    __shared__ float red[256];
    const int t = blockIdx.x;
    const int d = threadIdx.x;
    const float* row = qkv + (long)t * QKV_N;

    float qv[NH];
#pragma unroll
    for (int hh = 0; hh < NH; ++hh) qv[hh] = row[hh * HD + d];
    const float kv = row[NH * HD + d];
    const float vv = row[(NH + 1) * HD + d];

    float invq[NH];
#pragma unroll
    for (int hh = 0; hh < NH; ++hh)
        invq[hh] = rsqrtf(block_reduce_sum(qv[hh] * qv[hh], red) * (1.f / (float)HD) + 1e-6f);
    const float invk = rsqrtf(block_reduce_sum(kv * kv, red) * (1.f / (float)HD) + 1e-6f);

    const float cs = cosT[(long)t * HD + d];
    const float sn = sinT[(long)t * HD + d];

#pragma unroll
    for (int hh = 0; hh < NH; ++hh) {
        const float xn = qv[hh] * invq[hh] * (1.f + qw[d]);
        red[d] = xn;
        __syncthreads();
        const float rot = (d < HD / 2) ? -red[d + HD / 2] : red[d - HD / 2];
        qout[((long)hh * TQ + t) * HD + d] = xn * cs + rot * sn;
        __syncthreads();
    }

    const float kn = kv * invk * (1.f + kw[d]);
    red[d] = kn;
    __syncthreads();
    const float rotk = (d < HD / 2) ? -red[d + HD / 2] : red[d - HD / 2];
    const float kr = kn * cs + rotk * sn;

    kfull[(long)(CACHE_ + t) * HD + d] = kr;
    kslice[(long)t * HD + d]           = kr;
    vfull[(long)d * SK + CACHE_ + t]   = vv;
    vslice[(long)d * TQ + t]           = vv;
}

// scores row: softcap(tanh) + mask + softmax, in place.  grid = NH*T rows.
__global__ __launch_bounds__(256) void softcap_softmax_kernel(
    float* __restrict__ scores, const float* __restrict__ mask, float scale)
{
    __shared__ float red[256];
    const long row = blockIdx.x;
    const int  t   = (int)(row % TQ);
    float* sc = scores + row * SK;
    const float* mr = mask + (long)t * SK;

    float mx = -3.4e38f;
    for (int s = threadIdx.x; s < SK; s += 256) {
        const float v = tanhf(sc[s] * (scale / 50.0f)) * 50.0f + mr[s];
        sc[s] = v;
        mx = fmaxf(mx, v);
    }
    mx = block_reduce_max(mx, red);

    float sum = 0.f;
    for (int s = threadIdx.x; s < SK; s += 256) {
        const float e = expf(sc[s] - mx);
        sc[s] = e;
        sum += e;
    }
    const float inv = 1.f / block_reduce_sum(sum, red);
    for (int s = threadIdx.x; s < SK; s += 256) sc[s] *= inv;
}

// g = gelu_tanh(g) * u
__global__ __launch_bounds__(256) void gelu_mul_kernel(
    float* __restrict__ g, const float* __restrict__ u, long n)
{
    const long i = (long)blockIdx.x * 256 + threadIdx.x;
    if (i < n) {
        const float x = g[i];
        const float c = tanhf(0.7978845608028654f * (x + 0.044715f * x * x * x));
        g[i] = 0.5f * x * (1.f + c) * u[i];
    }
}

// ---------------------------------------------------------------------------
extern "C" void kernel_launch(void* const* d_in, const int* in_sizes, int n_in,
                              void* d_out, int out_size, void* d_ws, size_t ws_size,
                              hipStream_t stream)
{
    (void)in_sizes; (void)n_in; (void)out_size; (void)ws_size;

    const float* in_h    = (const float*)d_in[0];
    const float* mask_g  = (const float*)d_in[1];
    const float* mask_l  = (const float*)d_in[2];
    const float* cos_g   = (const float*)d_in[3];
    const float* sin_g   = (const float*)d_in[4];
    const float* cos_l   = (const float*)d_in[5];
    const float* sin_l   = (const float*)d_in[6];
    const float* cache_k = (const float*)d_in[7];
    const float* cache_v = (const float*)d_in[8];
    const float* pre_attn  = (const float*)d_in[9];
    const float* qnw       = (const float*)d_in[10];
    const float* knw       = (const float*)d_in[11];
    const float* qkv_w     = (const float*)d_in[12];
    const float* out_w     = (const float*)d_in[13];
    const float* post_attn = (const float*)d_in[14];
    const float* pre_ff    = (const float*)d_in[15];
    const float* gate_w    = (const float*)d_in[16];
    const float* up_w      = (const float*)d_in[17];
    const float* down_w    = (const float*)d_in[18];
    const float* post_ff   = (const float*)d_in[19];

    float* ws = (float*)d_ws;
    float* hbuf  = ws; ws += (long)TQ * D;
    float* xnorm = ws; ws += (long)TQ * D;
    float* qkvb  = ws; ws += (long)TQ * QKV_N;
    float* qb    = ws; ws += (long)NH * TQ * HD;
    float* kfull = ws; ws += (long)SK * HD;
    float* vfull = ws; ws += (long)HD * SK;
    float* scor  = ws; ws += (long)NH * TQ * SK;
    float* attn  = ws; ws += (long)TQ * AO;
    float* yb    = ws; ws += (long)TQ * D;
    float* gb    = ws; ws += (long)TQ * FFD;
    float* ub    = ws; ws += (long)TQ * FFD;

    float* out_h = (float*)d_out;
    float* out_k = out_h + (long)TQ * D;
    float* out_v = out_k + (long)LAYERS * TQ * HD;

    (void)hipMemcpyAsync(hbuf, in_h, sizeof(float) * (size_t)TQ * D,
                         hipMemcpyDeviceToDevice, stream);

    const float scale = 1.0f / 16.0f;  // 1/sqrt(HD)
    const int MB = TQ / 64;            // 8 M block-tiles (block covers 64 rows)

    for (int i = 0; i < LAYERS; ++i) {
        const bool  loc = ((i + 1) % 6) == 0;
        const float* cs = loc ? cos_l  : cos_g;
        const float* sn = loc ? sin_l  : sin_g;
        const float* mk = loc ? mask_l : mask_g;

        // ---- attention ----
        rmsnorm_kernel<<<TQ, 256, 0, stream>>>(hbuf, pre_attn + (long)i * D, xnorm, D);

        gemm_nn_t<D, D, QKV_N, QKV_N><<<dim3(QKV_N / 128, MB, 1), 256, 0, stream>>>(
            xnorm, qkv_w + (long)i * D * QKV_N, qkvb, 0, 0, 0);

        copy_rows_kernel<<<CACHE_, 256, 0, stream>>>(
            cache_k + (long)i * CACHE_ * HD, kfull, HD, HD, HD);
        copy_rows_kernel<<<HD, 256, 0, stream>>>(
            cache_v + (long)i * HD * CACHE_, vfull, CACHE_, CACHE_, SK);

        qkv_post_kernel<<<TQ, 256, 0, stream>>>(
            qkvb, qnw + (long)i * HD, knw + (long)i * HD, cs, sn,
            qb, kfull, vfull,
            out_k + (long)i * TQ * HD, out_v + (long)i * HD * TQ);

        // scores[h] = q[h] @ k_full^T     (M=T, N=S, K=HD), batched over heads
        gemm_nt_t<HD, HD, HD, SK><<<dim3(SK / 128, MB, NH), 256, 0, stream>>>(
            qb, kfull, scor, (long)TQ * HD, 0, (long)TQ * SK);

        softcap_softmax_kernel<<<NH * TQ, 256, 0, stream>>>(scor, mk, scale);

        // attn[:, h*HD:(h+1)*HD] = probs[h] @ v_full^T   (M=T, N=HD, K=S)
        gemm_nt_t<SK, SK, SK, AO><<<dim3(HD / 128, MB, NH), 256, 0, stream>>>(
            scor, vfull, attn, (long)TQ * SK, 0, (long)HD);

        gemm_nn_t<AO, AO, D, D><<<dim3(D / 128, MB, 1), 256, 0, stream>>>(
            attn, out_w + (long)i * AO * D, yb, 0, 0, 0);

        add_rmsnorm_kernel<<<TQ, 256, 0, stream>>>(hbuf, yb, post_attn + (long)i * D, D);

        // ---- MLP ----
        rmsnorm_kernel<<<TQ, 256, 0, stream>>>(hbuf, pre_ff + (long)i * D, xnorm, D);

        gemm_nn_t<D, D, FFD, FFD><<<dim3(FFD / 128, MB, 1), 256, 0, stream>>>(
            xnorm, gate_w + (long)i * D * FFD, gb, 0, 0, 0);
        gemm_nn_t<D, D, FFD, FFD><<<dim3(FFD / 128, MB, 1), 256, 0, stream>>>(
            xnorm, up_w + (long)i * D * FFD, ub, 0, 0, 0);

        const long nff = (long)TQ * FFD;
        gelu_mul_kernel<<<(unsigned)((nff + 255) / 256), 256, 0, stream>>>(gb, ub, nff);

        gemm_nn_t<FFD, FFD, D, D><<<dim3(D / 128, MB, 1), 256, 0, stream>>>(
            gb, down_w + (long)i * FFD * D, yb, 0, 0, 0);

        add_rmsnorm_kernel<<<TQ, 256, 0, stream>>>(hbuf, yb, post_ff + (long)i * D, D);
    }

    (void)hipMemcpyAsync(out_h, hbuf, sizeof(float) * (size_t)TQ * D,
                         hipMemcpyDeviceToDevice, stream);
}